// MemristorDense_22127671509681
// MI455X (gfx1250) — compile-verified
//
#include <hip/hip_runtime.h>
#include <math.h>

// ---------------------------------------------------------------------------
// MemristorDense forward for MI455X (gfx1250).
//
// y[b,j] = 0.5 * sum_i s[b,i] * ( (|wp|+c)*2^(ep*L) - (|wn|+c)*2^(en*L) )
//   L[b,i] = log2(2*|inputs[b,i]|), s = sign(inputs), e = log2(n_devices),
//   c = G_MIN * max|combined| / (G_MAX - G_MIN)
//
// Not expressible as WMMA (exponent couples (i,j) with (b,i) inside exp2),
// so this targets the VALU transcendental pipe + CDNA5 async/LDS data path.
// ---------------------------------------------------------------------------

#define NI   1024
#define NIP  1025
#define NO   512
#define NB   128
#define NOUT (NB * NO)     // 65536
#define KSPLIT 8
#define G_MIN_ 1e-5f
#define G_MAX_ 1e-3f

#define GLOBAL_AS __attribute__((address_space(1)))
#define LDS_AS    __attribute__((address_space(3)))

typedef int v2i __attribute__((ext_vector_type(2)));

#if __has_builtin(__builtin_amdgcn_global_load_async_to_lds_b64)
#define USE_ASYNC_LDS 1
#else
#define USE_ASYNC_LDS 0
#endif

__global__ __launch_bounds__(32) void k_init(unsigned* wsMax) {
    if (threadIdx.x == 0) *wsMax = __float_as_uint(0.5f); // bias |w| = 0.5 exactly
}

__global__ __launch_bounds__(256) void k_absmax(const float* __restrict__ a,
                                                const float* __restrict__ b,
                                                unsigned* wsMax) {
    __shared__ float red[256];
    const int N = NI * NO;
    float m = 0.f;
    for (int idx = blockIdx.x * 256 + threadIdx.x; idx < 2 * N; idx += gridDim.x * 256) {
        float v = (idx < N) ? a[idx] : b[idx - N];
        m = fmaxf(m, fabsf(v));
    }
    red[threadIdx.x] = m;
    __syncthreads();
    for (int s = 128; s > 0; s >>= 1) {
        if (threadIdx.x < s) red[threadIdx.x] = fmaxf(red[threadIdx.x], red[threadIdx.x + s]);
        __syncthreads();
    }
    // bitwise uint max == float max for non-negative floats; order-independent.
    if (threadIdx.x == 0) atomicMax(wsMax, __float_as_uint(red[0]));
}

__global__ __launch_bounds__(32) void k_scalar(const unsigned* wsMax, float* wsC) {
    if (threadIdx.x == 0) {
        float mw = __uint_as_float(*wsMax);
        *wsC = G_MIN_ * mw / (G_MAX_ - G_MIN_);
    }
}

// Precompute (L = log2(2|v|), s = sign(v)) per (b,i); i==NI is the bias input 1.
__global__ __launch_bounds__(256) void k_ls(const float* __restrict__ x, float2* __restrict__ LS) {
    int idx = blockIdx.x * 256 + threadIdx.x;
    if (idx >= NB * NIP) return;
    int b = idx / NIP;
    int i = idx - b * NIP;
    float v = (i < NI) ? x[b * NI + i] : 1.0f;
    float a = fabsf(v);
    float2 t;
    t.x = 1.0f + __builtin_amdgcn_logf(a);              // log2(2a); a==0 -> -inf, exp2 -> 0
    t.y = (v > 0.f) ? 1.0f : ((v < 0.f) ? -1.0f : 0.0f);
    LS[idx] = t;
}

// Main: grid (NO/32 j-tiles, NB/32 b-tiles, KSPLIT). 256 thr = 32 j-lanes x 8 b-groups,
// each thread owns 1 column-pair j and 4 batch rows (amortizes the 2 log2 per (i,j)).
__global__ __launch_bounds__(256) void k_main(const float* __restrict__ w_pos,
                                              const float* __restrict__ w_neg,
                                              const float* __restrict__ b_pos,
                                              const float* __restrict__ b_neg,
                                              const float* __restrict__ n_dev,
                                              const float* __restrict__ wsC,
                                              const float2* __restrict__ LS,
                                              float* __restrict__ partial) {
    const int t  = threadIdx.x;
    const int jl = t & 31;
    const int bg = t >> 5;
    const int j     = blockIdx.x * 32 + jl;
    const int bbase = blockIdx.y * 32;
    const int kt    = blockIdx.z;

    const int i0 = (kt * NIP) / KSPLIT;
    const int i1 = ((kt + 1) * NIP) / KSPLIT;

    const float c = *wsC;

    __shared__ float2 sLS[32][32];   // [i-chunk][batch row] — wave reads are same-address broadcasts

    float acc[4] = {0.f, 0.f, 0.f, 0.f};

    for (int ibase = i0; ibase < i1; ibase += 32) {
        int cnt = i1 - ibase;
        if (cnt > 32) cnt = 32;

        __syncthreads();            // previous chunk fully consumed
        for (int idx = t; idx < cnt * 32; idx += 256) {
            int ci = idx >> 5;
            int bb = idx & 31;
            const float2* src = &LS[(size_t)(bbase + bb) * NIP + (ibase + ci)];
#if USE_ASYNC_LDS
            __builtin_amdgcn_global_load_async_to_lds_b64(
                (GLOBAL_AS v2i*)src, (LDS_AS v2i*)&sLS[ci][bb], 0, 0);
#else
            sLS[ci][bb] = *src;
#endif
        }
#if USE_ASYNC_LDS
#if __has_builtin(__builtin_amdgcn_s_wait_asynccnt)
        __builtin_amdgcn_s_wait_asynccnt(0);
#else
        asm volatile("s_wait_asynccnt 0" ::: "memory");
#endif
#endif
        __syncthreads();

        for (int ci = 0; ci < cnt; ++ci) {
            const int i = ibase + ci;
            float2 n2 = *(const float2*)(n_dev + (size_t)i * (2 * NO) + 2 * j);
            float wp, wn;
            if (i < NI) {
                wp = w_pos[(size_t)i * NO + j];
                wn = w_neg[(size_t)i * NO + j];
            } else {
                wp = b_pos[j];
                wn = b_neg[j];
            }
            if (i + 32 < i1)   // pull next chunk's n_devices rows toward L2/L0
                __builtin_prefetch(n_dev + (size_t)(i + 32) * (2 * NO) + 2 * j, 0, 0);

            const float ep = __builtin_amdgcn_logf(n2.x);   // v_log_f32
            const float en = __builtin_amdgcn_logf(n2.y);
            const float ap = fabsf(wp) + c;
            const float an = fabsf(wn) + c;
#pragma unroll
            for (int r = 0; r < 4; ++r) {
                float2 ls = sLS[ci][bg * 4 + r];
                float tp = __builtin_amdgcn_exp2f(ep * ls.x);   // v_exp_f32
                float tn = __builtin_amdgcn_exp2f(en * ls.x);
                acc[r] = fmaf(ls.y, fmaf(ap, tp, -(an * tn)), acc[r]);
            }
        }
    }

#pragma unroll
    for (int r = 0; r < 4; ++r) {
        int b = bbase + bg * 4 + r;
        partial[(size_t)kt * NOUT + (size_t)b * NO + j] = 0.5f * acc[r];
    }
}

// Fixed-order split-K reduction -> deterministic output.
__global__ __launch_bounds__(256) void k_reduce(const float* __restrict__ partial,
                                                float* __restrict__ out) {
    int idx = blockIdx.x * 256 + threadIdx.x;
    if (idx >= NOUT) return;
    float s = 0.f;
    for (int k = 0; k < KSPLIT; ++k)
        s += partial[(size_t)k * NOUT + idx];
    out[idx] = s;
}

extern "C" void kernel_launch(void* const* d_in, const int* in_sizes, int n_in,
                              void* d_out, int out_size, void* d_ws, size_t ws_size,
                              hipStream_t stream) {
    const float* x     = (const float*)d_in[0];
    const float* w_pos = (const float*)d_in[1];
    const float* w_neg = (const float*)d_in[2];
    const float* b_pos = (const float*)d_in[3];
    const float* b_neg = (const float*)d_in[4];
    const float* n_dev = (const float*)d_in[5];
    float* out = (float*)d_out;

    // ws layout: [0..3] uint max bits | [4..7] float c | pad to 16B |
    //            LS float2[NB*NIP] | partial float[KSPLIT*NOUT]
    unsigned* wsMax = (unsigned*)d_ws;
    float*    wsC   = (float*)d_ws + 1;
    float2*   LS    = (float2*)((char*)d_ws + 16);
    float*    part  = (float*)((char*)d_ws + 16 + (size_t)NB * NIP * sizeof(float2));

    k_init<<<1, 32, 0, stream>>>(wsMax);
    k_absmax<<<512, 256, 0, stream>>>(w_pos, w_neg, wsMax);
    k_scalar<<<1, 32, 0, stream>>>(wsMax, wsC);
    k_ls<<<(NB * NIP + 255) / 256, 256, 0, stream>>>(x, LS);

    dim3 grid(NO / 32, NB / 32, KSPLIT);
    k_main<<<grid, 256, 0, stream>>>(w_pos, w_neg, b_pos, b_neg, n_dev, wsC, LS, part);

    k_reduce<<<(NOUT + 255) / 256, 256, 0, stream>>>(part, out);
}